// QuantizedCodebook_71459665871185
// MI455X (gfx1250) — compile-verified
//
#include <hip/hip_runtime.h>
#include <hip/hip_bf16.h>

typedef __attribute__((ext_vector_type(2))) float v2f;
typedef __attribute__((ext_vector_type(4))) float v4f;
typedef __attribute__((ext_vector_type(8))) float v8f;

#define DDIM 64
#define KCODES 1024
#define PHASE_ROWS 512          // codebook rows resident in LDS per phase
#define CB_STRIDE 68            // padded dword stride -> conflict-free ds_load_b64
#define WAVES 16
#define BLOCK (WAVES * 32)      // 512 threads
#define ROWS_PER_BLOCK (WAVES * 16)

__global__ __launch_bounds__(BLOCK)
void vq_wmma_kernel(const float* __restrict__ X, const float* __restrict__ C,
                    float* __restrict__ zq, float* __restrict__ out_idx,
                    float* __restrict__ block_partial)
{
    __shared__ float s_cb[PHASE_ROWS * CB_STRIDE];   // ~136 KB
    __shared__ float s_csqr[KCODES];                 // 4 KB
    __shared__ int   s_idx[WAVES][16];
    __shared__ float s_part[WAVES];

    const int tid  = threadIdx.x;
    const int w    = tid >> 5;
    const int lane = tid & 31;
    const int l16  = lane & 15;
    const int hlf  = lane >> 4;       // which 16-lane half
    const int kk   = hlf * 2;         // K sub-offset inside a K=4 chunk

    // ---- codebook squared norms (read from global; codebook is L2-resident) ----
    for (int r = tid; r < KCODES; r += BLOCK) {
        float s = 0.f;
        const float* row = C + r * DDIM;
        #pragma unroll
        for (int j = 0; j < DDIM / 4; ++j) {
            v4f q = *(const v4f*)(row + j * 4);
            s += q.x * q.x + q.y * q.y + q.z * q.z + q.w * q.w;
        }
        s_csqr[r] = s;
    }

    const int row_base = blockIdx.x * ROWS_PER_BLOCK + w * 16;
    const int arow     = row_base + l16;

    // ---- preload A fragments: 16 chunks of K=4 (lane holds K = kc*4 + kk + {0,1}) ----
    v2f a[16];
    #pragma unroll
    for (int kc = 0; kc < 16; ++kc)
        a[kc] = *(const v2f*)(X + (size_t)arow * DDIM + kc * 4 + kk);

    float minv[8];
    int   mini[8];
    #pragma unroll
    for (int v = 0; v < 8; ++v) { minv[v] = 3.4e38f; mini[v] = 0; }

    for (int phase = 0; phase < KCODES / PHASE_ROWS; ++phase) {
        __syncthreads();  // csqr ready / previous-phase consumers done
        // cooperative load of PHASE_ROWS codebook rows into padded LDS
        for (int i = tid; i < PHASE_ROWS * (DDIM / 4); i += BLOCK) {
            int r = i >> 4, s = i & 15;
            v4f q = *(const v4f*)(C + (size_t)(phase * PHASE_ROWS + r) * DDIM + s * 4);
            *(v4f*)(s_cb + r * CB_STRIDE + s * 4) = q;
        }
        __syncthreads();

        for (int ct = 0; ct < PHASE_ROWS / 16; ++ct) {
            const int nloc = ct * 16 + l16;               // this lane's code column
            const int n    = phase * PHASE_ROWS + nloc;
            // issue all B-fragment LDS loads up front (graded dscnt waits)
            v2f b[16];
            #pragma unroll
            for (int kc = 0; kc < 16; ++kc)               // B[k][n] = C[n][k]
                b[kc] = *(const v2f*)(s_cb + nloc * CB_STRIDE + kc * 4 + kk);
            // two independent accumulator chains -> half the WMMA RAW chain depth
            v8f c0 = {};
            v8f c1 = {};
            #pragma unroll
            for (int kc = 0; kc < 16; kc += 2) {
                c0 = __builtin_amdgcn_wmma_f32_16x16x4_f32(
                         false, a[kc],     false, b[kc],     (short)0, c0, false, false);
                c1 = __builtin_amdgcn_wmma_f32_16x16x4_f32(
                         false, a[kc + 1], false, b[kc + 1], (short)0, c1, false, false);
            }
            const float cs = s_csqr[n];
            #pragma unroll
            for (int v = 0; v < 8; ++v) {                 // dist = |c|^2 - 2 x.c
                float dot = c0[v] + c1[v];
                float d = __builtin_fmaf(-2.0f, dot, cs);
                if (d < minv[v]) { minv[v] = d; mini[v] = n; }
            }
        }
    }

    // ---- cross-lane argmin within each 16-lane half (rows v / 8+v) ----
    #pragma unroll
    for (int v = 0; v < 8; ++v) {
        float mv = minv[v]; int mi = mini[v];
        #pragma unroll
        for (int off = 1; off < 16; off <<= 1) {
            float ov = __shfl_xor(mv, off, 32);
            int   oi = __shfl_xor(mi, off, 32);
            if (ov < mv || (ov == mv && oi < mi)) { mv = ov; mi = oi; }
        }
        minv[v] = mv; mini[v] = mi;
    }
    if (l16 == 0) {
        #pragma unroll
        for (int v = 0; v < 8; ++v) s_idx[w][hlf * 8 + v] = mini[v];
    }
    __syncthreads();

    // ---- indices output (as float, matching concatenated-f32 output buffer) ----
    if (lane < 16) out_idx[row_base + lane] = (float)s_idx[w][lane];

    // ---- z_q = codebook gather; accumulate (x - q)^2 ----
    const int erow = lane >> 1;
    const int colb = (lane & 1) * 32;
    const int code = s_idx[w][erow];
    const float* xrow = X  + (size_t)(row_base + erow) * DDIM + colb;
    const float* qrow = C  + (size_t)code * DDIM + colb;
    float*       zrow = zq + (size_t)(row_base + erow) * DDIM + colb;
    float acc = 0.f;
    #pragma unroll
    for (int j = 0; j < 8; ++j) {
        v4f xv = *(const v4f*)(xrow + j * 4);
        v4f qv = *(const v4f*)(qrow + j * 4);
        *(v4f*)(zrow + j * 4) = qv;                 // z_q == quantize numerically
        v4f dd = xv - qv;
        acc += dd.x * dd.x + dd.y * dd.y + dd.z * dd.z + dd.w * dd.w;
    }
    #pragma unroll
    for (int off = 1; off < 32; off <<= 1) acc += __shfl_xor(acc, off, 32);
    if (lane == 0) s_part[w] = acc;
    __syncthreads();
    if (tid == 0) {                                 // fixed-order -> deterministic
        float s = 0.f;
        #pragma unroll
        for (int i = 0; i < WAVES; ++i) s += s_part[i];
        block_partial[blockIdx.x] = s;
    }
}

__global__ __launch_bounds__(256)
void vq_loss_reduce(const float* __restrict__ part, int nb, float scale,
                    float* __restrict__ out)
{
    __shared__ float s[256];
    float a = 0.f;
    for (int i = threadIdx.x; i < nb; i += 256) a += part[i];
    s[threadIdx.x] = a;
    __syncthreads();
    for (int o = 128; o > 0; o >>= 1) {
        if (threadIdx.x < o) s[threadIdx.x] += s[threadIdx.x + o];
        __syncthreads();
    }
    if (threadIdx.x == 0) out[0] = scale * s[0];
}

extern "C" void kernel_launch(void* const* d_in, const int* in_sizes, int n_in,
                              void* d_out, int out_size, void* d_ws, size_t ws_size,
                              hipStream_t stream)
{
    const float* X = (const float*)d_in[0];   // [N, 64] flattened inputs
    const float* C = (const float*)d_in[1];   // [1024, 64] codebook
    const int N = in_sizes[0] / DDIM;         // 131072

    float* out  = (float*)d_out;
    float* zq   = out + 1;                    // tuple order: loss, z_q, indices
    float* oidx = out + 1 + (size_t)N * DDIM;
    float* part = (float*)d_ws;

    const int nblocks = N / ROWS_PER_BLOCK;   // 512
    vq_wmma_kernel<<<nblocks, BLOCK, 0, stream>>>(X, C, zq, oidx, part);

    const float scale = 1.25f / (float)((size_t)N * DDIM);  // (1 + BETA) * mean
    vq_loss_reduce<<<1, 256, 0, stream>>>(part, nblocks, scale, out);
}